// CubicSpline_72834055406354
// MI455X (gfx1250) — compile-verified
//
#include <hip/hip_runtime.h>
#include <math.h>

typedef float v4f __attribute__((ext_vector_type(4)));

namespace {
constexpr int kKnots      = 128;
constexpr int kSeg        = 127;                  // kKnots - 1
constexpr int kCh         = 128;
constexpr int kCoefFloats = kSeg * 4 * kCh;       // 65024 floats
constexpr int kCoefBytes  = kCoefFloats * 4;      // 260096 B
constexpr int kKnotBytes  = kKnots * 4;           // 512 B
constexpr int kLdsFloats  = kCoefFloats + kKnots; // 260608 B total LDS
constexpr int kBlock      = 1024;                 // 32 wave32 = 8 waves/SIMD
constexpr int kGrid       = 512;
constexpr int kBatch      = 32;                   // trials per wave batch
}

__device__ __forceinline__ void eval_trial(const float* __restrict__ lds,
                                           const float* __restrict__ knots,
                                           float r, float knot0, float inv_h,
                                           float rmax, int lane,
                                           float* __restrict__ dst)
{
    int idx = (int)((r - knot0) * inv_h);          // r >= 0: trunc == floor
    idx = idx < 0 ? 0 : idx;
    idx = idx > (kSeg - 1) ? (kSeg - 1) : idx;
    float dx = r - knots[idx];

    const v4f* cbase = (const v4f*)&lds[(unsigned)idx * (4 * kCh) + (lane << 2)];
    v4f a  = cbase[0];                 // coef[idx][0][ch..ch+3]
    v4f b  = cbase[kCh / 4];           // coef[idx][1][...]
    v4f cc = cbase[2 * (kCh / 4)];     // coef[idx][2][...]
    v4f d  = cbase[3 * (kCh / 4)];     // coef[idx][3][...]

    v4f o = a + dx * (b + dx * (cc + dx * d));     // -> v_fma
    v4f zero = {0.0f, 0.0f, 0.0f, 0.0f};
    o = (r < rmax) ? o : zero;

    __builtin_nontemporal_store(o, (v4f*)dst);
}

__launch_bounds__(kBlock, 1)
__global__ void spline_eval_kernel(const float* __restrict__ r_trial,
                                   const float* __restrict__ r_knots,
                                   const float* __restrict__ coef,
                                   const float* __restrict__ h_ptr,
                                   const float* __restrict__ rmax_ptr,
                                   float* __restrict__ out,
                                   int n_trial)
{
    __shared__ float lds[kLdsFloats];
    const unsigned lds_base = (unsigned)(size_t)(void*)&lds[0];
    const int tid = threadIdx.x;

    // ---- Stage coefficient table + knots into LDS via CDNA5 async copies
    // (ASYNCcnt-tracked, 16 B per lane per issue, fully coalesced).
    for (int chunk = tid; chunk < kCoefBytes / 16; chunk += kBlock) {
        unsigned dst = lds_base + (unsigned)(chunk * 16);
        const void* src = (const char*)coef + (size_t)chunk * 16;
        asm volatile("global_load_async_to_lds_b128 %0, %1, off"
                     :: "v"(dst), "v"(src) : "memory");
    }
    for (int chunk = tid; chunk < kKnotBytes / 16; chunk += kBlock) {
        unsigned dst = lds_base + (unsigned)(kCoefBytes + chunk * 16);
        const void* src = (const char*)r_knots + (size_t)chunk * 16;
        asm volatile("global_load_async_to_lds_b128 %0, %1, off"
                     :: "v"(dst), "v"(src) : "memory");
    }
    asm volatile("s_wait_asynccnt 0x0" ::: "memory");
    __syncthreads();

    const float h      = h_ptr[0];
    const float rmax   = rmax_ptr[0];
    const float inv_h  = 1.0f / h;
    const float* knots = &lds[kCoefFloats];
    const float knot0  = knots[0];

    const int  lane   = tid & 31;                       // wave32
    const long wid    = (long)blockIdx.x * (kBlock / 32) + (tid >> 5);
    const long nwaves = (long)gridDim.x * (kBlock / 32);
    const long n      = (long)n_trial;

    // Each wave owns batches of 32 consecutive trials. One coalesced 128 B
    // load pulls 32 r-values; __shfl (ds_bpermute) broadcasts them so the
    // global-load latency is off the per-trial critical path. Per trial:
    // 32 lanes x 4 channels (float4) = 128 ch -> one contiguous 512 B NT
    // store. Fast path has no per-trial bounds checks.
    for (long base = wid * kBatch; base < n; base += nwaves * kBatch) {
        if (base + kBatch <= n) {                        // full batch (common)
            float rv = __builtin_nontemporal_load(&r_trial[base + lane]);
            float* obase = out + (size_t)base * kCh + (lane << 2);
            #pragma unroll 4
            for (int j = 0; j < kBatch; ++j) {
                float r = __shfl(rv, j, 32);
                eval_trial(lds, knots, r, knot0, inv_h, rmax, lane,
                           obase + (size_t)j * kCh);
            }
        } else {                                         // ragged tail
            long tl = base + lane;
            float rv = __builtin_nontemporal_load(&r_trial[tl < n ? tl : (n - 1)]);
            for (int j = 0; j < kBatch; ++j) {
                long t = base + j;
                if (t < n) {                             // wave-uniform guard
                    float r = __shfl(rv, j, 32);
                    eval_trial(lds, knots, r, knot0, inv_h, rmax, lane,
                               out + (size_t)t * kCh + (lane << 2));
                }
            }
        }
    }
}

extern "C" void kernel_launch(void* const* d_in, const int* in_sizes, int n_in,
                              void* d_out, int out_size, void* d_ws, size_t ws_size,
                              hipStream_t stream) {
    const float* r_trial = (const float*)d_in[0];
    const float* r_knots = (const float*)d_in[1];
    const float* coef    = (const float*)d_in[2];
    const float* h       = (const float*)d_in[3];
    const float* rmax    = (const float*)d_in[4];
    float* out = (float*)d_out;
    const int n = in_sizes[0];

    spline_eval_kernel<<<dim3(kGrid), dim3(kBlock), 0, stream>>>(
        r_trial, r_knots, coef, h, rmax, out, n);
}